// LightGCN_71124658422236
// MI455X (gfx1250) — compile-verified
//
#include <hip/hip_runtime.h>
#include <hip/hip_bf16.h>

typedef __attribute__((ext_vector_type(2))) float v2f;
typedef __attribute__((ext_vector_type(8))) float v8f;

#define DIM 64

// ---------------------------------------------------------------------------
// Zero workspace (poisoned with 0xAA by harness; we need zeros for atomics).
// float4 stores, memory-bandwidth bound.
// ---------------------------------------------------------------------------
__global__ void zero_ws_kernel(float4* __restrict__ p, long long n4) {
    long long i = (long long)blockIdx.x * blockDim.x + threadIdx.x;
    if (i < n4) p[i] = make_float4(0.f, 0.f, 0.f, 0.f);
}

// ---------------------------------------------------------------------------
// COO SpMM: dst[e_dst[e]] += vals[e] * src[e_src[e]]   (rows of DIM=64 f32)
// 16 threads per edge, each handling one float4 chunk (coalesced gather,
// hardware global_atomic_add_f32 scatter).
// ---------------------------------------------------------------------------
__global__ __launch_bounds__(256) void spmm_edges_kernel(
    const float* __restrict__ src, float* __restrict__ dst,
    const int* __restrict__ e_src, const int* __restrict__ e_dst,
    const float* __restrict__ vals, long long n_edges) {
    long long gid = (long long)blockIdx.x * blockDim.x + threadIdx.x;
    long long e = gid >> 4;        // edge index
    int q = (int)(gid & 15);       // which float4 of the 64-dim row
    if (e >= n_edges) return;
    int s = e_src[e];
    int d = e_dst[e];
    float v = vals[e];
    const float4 x = *(const float4*)(src + (long long)s * DIM + q * 4);
    float* out = dst + (long long)d * DIM + q * 4;
    unsafeAtomicAdd(out + 0, v * x.x);
    unsafeAtomicAdd(out + 1, v * x.y);
    unsafeAtomicAdd(out + 2, v * x.z);
    unsafeAtomicAdd(out + 3, v * x.w);
}

// ---------------------------------------------------------------------------
// fin = (e0 + l1 + l2) / 3, written in place over l1.
// ---------------------------------------------------------------------------
__global__ void combine3_kernel(const float4* __restrict__ e0,
                                float4* __restrict__ l1,
                                const float4* __restrict__ l2, long long n4) {
    long long i = (long long)blockIdx.x * blockDim.x + threadIdx.x;
    if (i >= n4) return;
    float4 a = e0[i], b = l1[i], c = l2[i];
    const float s = (1.0f / 3.0f);
    float4 r;
    r.x = (a.x + b.x + c.x) * s;
    r.y = (a.y + b.y + c.y) * s;
    r.z = (a.z + b.z + c.z) * s;
    r.w = (a.w + b.w + c.w) * s;
    l1[i] = r;
}

// ---------------------------------------------------------------------------
// Scoring with V_WMMA_F32_16X16X4_F32 (full f32 precision).
// One wave handles 16 batch elements. A = 16 rows of u_fin (16x4 per K-step),
// B[k][n] = i_fin[item[n]][k] (4x16). D = A*B^T-style product; diagonal
// D[b][b] = dot(u_fin[user[b]], i_fin[item[b]]). 16 K-steps cover DIM=64.
// Two accumulators (pos/neg) share the A operand.
//
// A layout  (16x4 f32, 2 VGPRs): lane L -> m = L&15, k = 2*(L>>4) + {0,1}
// B layout  (4x16 f32, 2 VGPRs): lane L -> n = L&15, k = 2*(L>>4) + {0,1}
// C/D layout(16x16 f32, 8 VGPRs): vgpr v, lanes 0-15 -> M=v, N=lane;
//                                 lanes 16-31 -> M=v+8, N=lane-16.
// Diagonal: lanes 0..7 give rows 0..7 (vgpr=lane);
//           lanes 16..23 write rows 8..15 from vgpr=lane-16, i.e. D[lane-8][lane-16] (off-diagonal).
// ---------------------------------------------------------------------------
__global__ __launch_bounds__(256) void score_wmma_kernel(
    const float* __restrict__ ufin, const float* __restrict__ ifin,
    const int* __restrict__ user, const int* __restrict__ pos_item,
    const int* __restrict__ neg_item, float* __restrict__ out, int batch) {
    const int lane = threadIdx.x & 31;
    const int wave = threadIdx.x >> 5;
    const int tile = blockIdx.x * 8 + wave;   // 16 batch elements per tile
    const int m = lane & 15;                  // row for A / col for B
    const int hi = lane >> 4;                 // selects K pair within K-step
    const int b = tile * 16 + m;

    const float* __restrict__ urow = ufin + (long long)user[b] * DIM;
    const float* __restrict__ prow = ifin + (long long)pos_item[b] * DIM;
    const float* __restrict__ nrow = ifin + (long long)neg_item[b] * DIM;

    v8f cp = {};
    v8f cn = {};
#pragma unroll
    for (int kk = 0; kk < DIM / 4; ++kk) {
        const int kb = kk * 4 + hi * 2;
        v2f a;  a.x  = urow[kb]; a.y  = urow[kb + 1];
        v2f bp; bp.x = prow[kb]; bp.y = prow[kb + 1];
        v2f bn; bn.x = nrow[kb]; bn.y = nrow[kb + 1];
        // 8-arg form: (neg_a, A, neg_b, B, c_mod, C, reuse_a, reuse_b)
        cp = __builtin_amdgcn_wmma_f32_16x16x4_f32(false, a, false, bp,
                                                   (short)0, cp, false, false);
        cn = __builtin_amdgcn_wmma_f32_16x16x4_f32(false, a, false, bn,
                                                   (short)0, cn, false, false);
    }

    // Diagonal extraction + store (divergence is fine after the WMMAs).
    if (lane < 8) {
        out[tile * 16 + lane]         = cp[lane];
        out[batch + tile * 16 + lane] = cn[lane];
    } else if (lane >= 16 && lane < 24) {
        const int row = lane - 8;      // rows 8..15
        const int v   = lane - 16;     // accumulator vgpr index
        out[tile * 16 + row]         = cp[v];
        out[batch + tile * 16 + row] = cn[v];
    }
}

// ---------------------------------------------------------------------------
// Host-side orchestration.
// Inputs: 0 user_emb [U*64] f32, 1 item_emb [I*64] f32, 2 edge_vals [E] f32,
//         3 edge_user [E] i32, 4 edge_item [E] i32,
//         5 user [B] i32, 6 pos_item [B] i32, 7 neg_item [B] i32.
// Output: pos_score [B] then neg_score [B], f32.
// Workspace layout (floats): i1 | u1 | i2 | u2.
// ---------------------------------------------------------------------------
extern "C" void kernel_launch(void* const* d_in, const int* in_sizes, int n_in,
                              void* d_out, int out_size, void* d_ws, size_t ws_size,
                              hipStream_t stream) {
    const float* user_emb = (const float*)d_in[0];
    const float* item_emb = (const float*)d_in[1];
    const float* edge_vals = (const float*)d_in[2];
    const int* edge_user = (const int*)d_in[3];
    const int* edge_item = (const int*)d_in[4];
    const int* user = (const int*)d_in[5];
    const int* pos_item = (const int*)d_in[6];
    const int* neg_item = (const int*)d_in[7];
    float* out = (float*)d_out;

    const long long U = in_sizes[0] / DIM;
    const long long I = in_sizes[1] / DIM;
    const long long E = in_sizes[2];
    const int B = in_sizes[5];

    float* ws = (float*)d_ws;
    float* i1 = ws;
    float* u1 = i1 + I * DIM;
    float* i2 = u1 + U * DIM;
    float* u2 = i2 + I * DIM;
    const long long ws_floats = 2 * (U + I) * DIM;

    // 1) zero the accumulation buffers
    {
        long long n4 = ws_floats / 4;
        int blocks = (int)((n4 + 255) / 256);
        zero_ws_kernel<<<blocks, 256, 0, stream>>>((float4*)ws, n4);
    }

    const long long spmm_threads = E * 16;
    const int spmm_blocks = (int)((spmm_threads + 255) / 256);

    // 2) i1 = A^T u0  (users -> items)
    spmm_edges_kernel<<<spmm_blocks, 256, 0, stream>>>(
        user_emb, i1, edge_user, edge_item, edge_vals, E);
    // 3) u1 = A i0    (items -> users)
    spmm_edges_kernel<<<spmm_blocks, 256, 0, stream>>>(
        item_emb, u1, edge_item, edge_user, edge_vals, E);
    // 4) i2 = A^T u1
    spmm_edges_kernel<<<spmm_blocks, 256, 0, stream>>>(
        u1, i2, edge_user, edge_item, edge_vals, E);
    // 5) u2 = A i1
    spmm_edges_kernel<<<spmm_blocks, 256, 0, stream>>>(
        i1, u2, edge_item, edge_user, edge_vals, E);

    // 6) u_fin -> u1 ; i_fin -> i1
    {
        long long nu4 = U * DIM / 4;
        int bu = (int)((nu4 + 255) / 256);
        combine3_kernel<<<bu, 256, 0, stream>>>((const float4*)user_emb,
                                                (float4*)u1, (const float4*)u2, nu4);
        long long ni4 = I * DIM / 4;
        int bi = (int)((ni4 + 255) / 256);
        combine3_kernel<<<bi, 256, 0, stream>>>((const float4*)item_emb,
                                                (float4*)i1, (const float4*)i2, ni4);
    }

    // 7) WMMA scoring: 16 batch elems / wave, 8 waves / block
    {
        int tiles = B / 16;               // 1024
        int blocks = tiles / 8;           // 128
        score_wmma_kernel<<<blocks, 256, 0, stream>>>(u1, i1, user, pos_item,
                                                      neg_item, out, B);
    }
}